// GradientConv_17824114278647
// MI455X (gfx1250) — compile-verified
//
#include <hip/hip_runtime.h>
#include <hip/hip_bf16.h>
#include <stdint.h>

#define CH   128          // channels
#define CD   256          // channels * D (D = 2)
#define NW   8            // waves per block (wave32)
#define WPT  4            // edges per wave, double-buffered
#define EPB  (NW * WPT)   // edges per block = 32

typedef unsigned int v4u __attribute__((ext_vector_type(4)));
typedef int          v8i __attribute__((ext_vector_type(8)));
typedef int          v4i __attribute__((ext_vector_type(4)));

__global__ void zero_out_kernel(float* __restrict__ o, int n) {
    int i4 = (blockIdx.x * blockDim.x + threadIdx.x) * 4;
    if (i4 + 3 < n) {
        float4 z; z.x = 0.f; z.y = 0.f; z.z = 0.f; z.w = 0.f;
        *(float4*)(o + i4) = z;
    } else {
        for (int j = i4; j < n; ++j) o[j] = 0.f;
    }
}

// 1-D TDM copy: n_dwords valid elements (zero-filled to a 64-dword tile) into LDS.
// D# per CDNA5 ISA 8.3/8.4: group0 = {count/type/lds/global addr}, group1 = dims.
__device__ __forceinline__ void tdm_load_1d(uint32_t lds_addr, const void* gptr, int n_dwords) {
    const uint64_t ga = (uint64_t)(uintptr_t)gptr;
    v4u g0;
    g0.x = 1u;                                                  // count=1, user desc, no gather
    g0.y = lds_addr;                                            // lds_addr (bytes)
    g0.z = (uint32_t)ga;                                        // global_addr[31:0]
    g0.w = ((uint32_t)(ga >> 32) & 0x01FFFFFFu) | 0x80000000u;  // global_addr[56:32] | type=2
    v8i g1;
    g1[0] = (2 << 16);                                          // data_size = 4 bytes
    g1[1] = (n_dwords & 0xFFFF) << 16;                          // tensor_dim0[15:0]
    g1[2] = ((n_dwords >> 16) & 0xFFFF) | (1 << 16);            // tensor_dim0[31:16], tensor_dim1=1
    g1[3] = (64 << 16);                                         // tile_dim0 = 64 dwords
    g1[4] = 0;                                                  // tile_dim1=0 (1-D), tile_dim2=0
    g1[5] = 64;                                                 // tensor_dim0_stride (unused, 1-D)
    g1[6] = 0;
    g1[7] = 0;
    v4i z4; z4.x = 0; z4.y = 0; z4.z = 0; z4.w = 0;             // groups 2/3: dims 2..4 unused
    v8i z8;
    z8[0] = 0; z8[1] = 0; z8[2] = 0; z8[3] = 0;
    z8[4] = 0; z8[5] = 0; z8[6] = 0; z8[7] = 0;
    // clang-23 / amdgpu-toolchain 6-arg form.
    __builtin_amdgcn_tensor_load_to_lds(g0, g1, z4, z4, z8, 0);
}

__global__ __launch_bounds__(NW * 32)
void GradientConv_17824114278647_kernel(const float* __restrict__ x,
                                        const float* __restrict__ ew,
                                        const int*   __restrict__ de,
                                        float*       __restrict__ out,
                                        int E) {
    // [wave][slot(2)][row(s,t)][CH] floats = 16 KB row staging
    __shared__ float  sh[NW * 2 * 2 * CH];
    __shared__ int2   de_t[EPB];    // TDM-staged edge indices (target, source)
    __shared__ float2 ew_t[EPB];    // TDM-staged edge weights

    const int wave  = threadIdx.x >> 5;
    const int lane  = threadIdx.x & 31;
    const int tile0 = blockIdx.x * EPB;
    const int ebase = tile0 + wave * WPT;

    // --- Stage the block's edge metadata with the Tensor Data Mover ---
    if (wave == 0) {
        const int rem   = E - tile0;
        const int valid = rem < EPB ? rem : EPB;
        tdm_load_1d((uint32_t)(uintptr_t)de_t, de + 2 * (size_t)tile0, valid * 2);
        tdm_load_1d((uint32_t)(uintptr_t)ew_t, ew + 2 * (size_t)tile0, valid * 2);
        __builtin_amdgcn_s_wait_tensorcnt(0);
    }
    __syncthreads();

    float* shw = sh + wave * (2 * 2 * CH);
    const uint32_t lds_base = (uint32_t)(uintptr_t)shw;   // low 32 bits of flat = LDS offset

    int   tk[WPT];
    float w0k[WPT], w1k[WPT];

    auto issue = [&](int k, int slot) {
        const int kidx = wave * WPT + k;
        const int2   st = de_t[kidx];      // {target, source}
        const float2 w  = ew_t[kidx];
        tk[k] = st.x; w0k[k] = w.x; w1k[k] = w.y;

        const uint64_t gs = (uint64_t)(uintptr_t)(x + (size_t)st.y * CH) + (uint32_t)(lane * 16);
        const uint64_t gt = (uint64_t)(uintptr_t)(x + (size_t)st.x * CH) + (uint32_t)(lane * 16);
        const uint32_t ls = lds_base + (uint32_t)slot * (2u * CH * 4u) + (uint32_t)(lane * 16);
        const uint32_t lt = ls + CH * 4u;

        // Ensure any prior DS reads of this (reused) slot have executed before
        // the async unit can overwrite it.
        asm volatile("s_wait_dscnt 0x0" ::: "memory");
        // DMA one 512B node row per instruction into LDS (16B per lane, wave32).
        asm volatile("global_load_async_to_lds_b128 %0, %1, off"
                     :: "v"(ls), "v"(gs) : "memory");
        asm volatile("global_load_async_to_lds_b128 %0, %1, off"
                     :: "v"(lt), "v"(gt) : "memory");
        // Stream-ahead hint for the next block's edge tile (global_prefetch_b8).
        __builtin_prefetch(de + 2 * (size_t)(tile0 + EPB), 0, 0);
    };

    auto compute = [&](int k, int slot) {
        const float wsel = (lane & 1) ? w1k[k] : w0k[k];
        const float* so  = shw + slot * (2 * CH);   // source row
        const float* to  = so + CH;                 // target row
        float* orow = out + (size_t)tk[k] * CD;
        #pragma unroll
        for (int i = 0; i < 8; ++i) {
            const int c = i * 16 + (lane >> 1);     // pairs of lanes broadcast one bank
            const float diff = so[c] - to[c];
            // out[t*256 + c*2 + d] with d = lane&1  ==  orow[i*32 + lane]
            // -> each atomic instruction touches one contiguous 128B line.
            atomicAdd(orow + i * 32 + lane, wsel * diff);
        }
    };

    if (ebase < E) issue(0, 0);
    #pragma unroll
    for (int k = 0; k < WPT; ++k) {
        const bool more = (k + 1 < WPT) && (ebase + k + 1 < E);
        if (more) {
            issue(k + 1, (k + 1) & 1);
            // stage k's 2 async loads complete (async loads retire in order);
            // stage k+1's 2 remain in flight.
            asm volatile("s_wait_asynccnt 0x2" ::: "memory");
        } else {
            asm volatile("s_wait_asynccnt 0x0" ::: "memory");
        }
        if (ebase + k < E) compute(k, k & 1);
    }
}

extern "C" void kernel_launch(void* const* d_in, const int* in_sizes, int n_in,
                              void* d_out, int out_size, void* d_ws, size_t ws_size,
                              hipStream_t stream) {
    const float* x  = (const float*)d_in[0];   // [1, N, 128] f32
    const float* ew = (const float*)d_in[1];   // [1, E, 2]   f32
    const int*   de = (const int*)d_in[2];     // [1, E, 2]   int32 (target, source)
    float* out = (float*)d_out;                // [1, N, 256] f32

    const int E = in_sizes[2] / 2;

    // Zero the scatter accumulator (harness poisons d_out).
    {
        int nthr = (out_size + 3) / 4;
        int blocks = (nthr + 255) / 256;
        zero_out_kernel<<<blocks, 256, 0, stream>>>(out, out_size);
    }
    // Gather-multiply-scatter.
    {
        int blocks = (E + EPB - 1) / EPB;
        GradientConv_17824114278647_kernel<<<blocks, NW * 32, 0, stream>>>(x, ew, de, out, E);
    }
}